// GCNII_19344532701768
// MI455X (gfx1250) — compile-verified
//
#include <hip/hip_runtime.h>
#include <cmath>

#define NN   50000
#define FIN  128
#define HH   128
#define CC   40
#define LL   4
#define EE   800000

typedef __attribute__((ext_vector_type(2))) float v2f;
typedef __attribute__((ext_vector_type(8))) float v8f;

// ---------------------------------------------------------------- degrees ---
__global__ void k_init_deg(float* __restrict__ deg) {
  int i = blockIdx.x * blockDim.x + threadIdx.x;
  if (i < NN) deg[i] = 1.0f;                       // self-loop contributes 1
}

__global__ void k_accum_deg(const int* __restrict__ ei, float* __restrict__ deg) {
  int e = blockIdx.x * blockDim.x + threadIdx.x;
  if (e < EE) {
    int dst = ei[e];                               // edge_index[0] = row (dest)
    __hip_atomic_fetch_add(&deg[dst], 1.0f, __ATOMIC_RELAXED,
                           __HIP_MEMORY_SCOPE_AGENT);
  }
}

__global__ void k_finalize_dinv(float* __restrict__ dinv) {
  int i = blockIdx.x * blockDim.x + threadIdx.x;
  if (i < NN) {
    float d = dinv[i];
    dinv[i] = d > 0.0f ? rsqrtf(d) : 0.0f;
  }
}

// --------------------------------------------------------- graph aggregate ---
// agg[i][:] = dinv[i]^2 * h[i][:]   (self-loop term), float4 per thread
__global__ void k_agg_init(const float* __restrict__ h,
                           const float* __restrict__ dinv,
                           float* __restrict__ agg) {
  int i = blockIdx.x * blockDim.x + threadIdx.x;   // float4 index
  if (i < NN * HH / 4) {
    int node = i >> 5;                             // 32 float4 per node (H=128)
    float c = dinv[node]; c = c * c;
    float4 v = ((const float4*)h)[i];
    float4 o; o.x = c*v.x; o.y = c*v.y; o.z = c*v.z; o.w = c*v.w;
    ((float4*)agg)[i] = o;
  }
}

// one wave32 per edge; lane covers 4 features -> 128 features
__global__ void k_edge_scatter(const int* __restrict__ ei,
                               const float* __restrict__ dinv,
                               const float* __restrict__ h,
                               float* __restrict__ agg) {
  int gid  = blockIdx.x * blockDim.x + threadIdx.x;
  int e    = gid >> 5;
  int lane = threadIdx.x & 31;
  if (e >= EE) return;
  int dst = ei[e];                                 // row (destination)
  int src = ei[EE + e];                            // col (source)
  float c = dinv[dst] * dinv[src];
  float4 v = *(const float4*)(h + (size_t)src * HH + lane * 4);
  float* a = agg + (size_t)dst * HH + lane * 4;
  __hip_atomic_fetch_add(a + 0, c * v.x, __ATOMIC_RELAXED, __HIP_MEMORY_SCOPE_AGENT);
  __hip_atomic_fetch_add(a + 1, c * v.y, __ATOMIC_RELAXED, __HIP_MEMORY_SCOPE_AGENT);
  __hip_atomic_fetch_add(a + 2, c * v.z, __ATOMIC_RELAXED, __HIP_MEMORY_SCOPE_AGENT);
  __hip_atomic_fetch_add(a + 3, c * v.w, __ATOMIC_RELAXED, __HIP_MEMORY_SCOPE_AGENT);
}

// agg = 0.9*agg + 0.1*h0   (support), in place
__global__ void k_support(float* __restrict__ agg, const float* __restrict__ h0) {
  int i = blockIdx.x * blockDim.x + threadIdx.x;
  if (i < NN * HH / 4) {
    float4 a = ((const float4*)agg)[i];
    float4 z = ((const float4*)h0)[i];
    float4 o;
    o.x = 0.9f*a.x + 0.1f*z.x;  o.y = 0.9f*a.y + 0.1f*z.y;
    o.z = 0.9f*a.z + 0.1f*z.z;  o.w = 0.9f*a.w + 0.1f*z.w;
    ((float4*)agg)[i] = o;
  }
}

// ------------------------------------------------------------------ WMMA ----
// Cooperative stage of the contiguous 16x128 A tile (8 KB) into LDS.
__device__ __forceinline__ void stage_A(const float* __restrict__ Ag, float* As) {
  for (int i = threadIdx.x; i < 16 * 128 / 4; i += blockDim.x)
    ((float4*)As)[i] = ((const float4*)Ag)[i];
  __syncthreads();
}

// One wave computes a 16x16 f32 tile of A(16x128)*B(128x16) via
// 32x V_WMMA_F32_16X16X4_F32; A from LDS, B (weights) from global/L2.
__device__ __forceinline__ v8f wmma_kloop128(const float* __restrict__ As,
                                             const float* __restrict__ B, int ldb,
                                             int l16, int half) {
  v8f acc = {};
#pragma unroll
  for (int k = 0; k < 128; k += 4) {
    int ka = k + 2 * half;
    v2f a, b;
    a.x = As[l16 * 128 + ka];         // A: M = l16, K pair in VGPR0/1 (LDS)
    a.y = As[l16 * 128 + ka + 1];
    b.x = B[ka * ldb + l16];          // B: N = l16, K pair per half-wave
    b.y = B[(ka + 1) * ldb + l16];
    acc = __builtin_amdgcn_wmma_f32_16x16x4_f32(
        /*neg_a=*/false, a, /*neg_b=*/false, b,
        /*c_mod=*/(short)0, acc, /*reuse_a=*/false, /*reuse_b=*/false);
  }
  return acc;
}

// h = h0 = relu(x @ W_in + b_in); grid = 3125 row tiles, 8 waves = 8 col tiles
__global__ void k_gemm_in(const float* __restrict__ x, const float* __restrict__ Win,
                          const float* __restrict__ bin, float* __restrict__ h,
                          float* __restrict__ h0) {
  __shared__ float As[16 * 128];
  int mt = blockIdx.x, nt = threadIdx.x >> 5;
  int lane = threadIdx.x & 31, half = lane >> 4, l16 = lane & 15;
  int row0 = mt * 16, col0 = nt * 16;
  stage_A(x + (size_t)row0 * FIN, As);
  v8f acc = wmma_kloop128(As, Win + col0, HH, l16, half);
  float bias = bin[col0 + l16];
#pragma unroll
  for (int r = 0; r < 8; ++r) {
    int row = row0 + r + 8 * half;
    float v = acc[r] + bias;
    v = v > 0.0f ? v : 0.0f;
    size_t idx = (size_t)row * HH + col0 + l16;
    h[idx] = v; h0[idx] = v;
  }
}

// h = relu(beta*(S @ W) + (1-beta)*S); S tile re-read from LDS
__global__ void k_gemm_layer(const float* __restrict__ sup, const float* __restrict__ W,
                             float* __restrict__ h, float beta) {
  __shared__ float As[16 * 128];
  int mt = blockIdx.x, nt = threadIdx.x >> 5;
  int lane = threadIdx.x & 31, half = lane >> 4, l16 = lane & 15;
  int row0 = mt * 16, col0 = nt * 16;
  stage_A(sup + (size_t)row0 * HH, As);
  v8f acc = wmma_kloop128(As, W + col0, HH, l16, half);
  float omb = 1.0f - beta;
#pragma unroll
  for (int r = 0; r < 8; ++r) {
    int row = row0 + r + 8 * half;
    float s = As[(r + 8 * half) * 128 + col0 + l16];   // S from LDS
    float v = beta * acc[r] + omb * s;
    h[(size_t)row * HH + col0 + l16] = v > 0.0f ? v : 0.0f;
  }
}

// out = h @ W_out + b_out  (C=40: 3 col tiles, value-masked past col 39)
__global__ void k_gemm_out(const float* __restrict__ h, const float* __restrict__ Wout,
                           const float* __restrict__ bout, float* __restrict__ out) {
  __shared__ float As[16 * 128];
  int mt = blockIdx.x, nt = threadIdx.x >> 5;
  int row0 = mt * 16;
  stage_A(h + (size_t)row0 * HH, As);                  // all 4 waves help
  if (nt >= 3) return;                                 // wave-uniform exit
  int lane = threadIdx.x & 31, half = lane >> 4, l16 = lane & 15;
  int col0 = nt * 16;
  int n = col0 + l16;
  float msk = (n < CC) ? 1.0f : 0.0f;
  int nc = (n < CC) ? n : 0;                           // clamp address, mask value
  v8f acc = {};
#pragma unroll
  for (int k = 0; k < 128; k += 4) {
    int ka = k + 2 * half;
    v2f a, b;
    a.x = As[l16 * 128 + ka];
    a.y = As[l16 * 128 + ka + 1];
    b.x = msk * Wout[ka * CC + nc];
    b.y = msk * Wout[(ka + 1) * CC + nc];
    acc = __builtin_amdgcn_wmma_f32_16x16x4_f32(false, a, false, b,
                                                (short)0, acc, false, false);
  }
  float bias = msk * bout[nc];
  if (n < CC) {
#pragma unroll
    for (int r = 0; r < 8; ++r) {
      int row = row0 + r + 8 * half;
      out[(size_t)row * CC + n] = acc[r] + bias;
    }
  }
}

// ------------------------------------------------------------------ launch ---
extern "C" void kernel_launch(void* const* d_in, const int* in_sizes, int n_in,
                              void* d_out, int out_size, void* d_ws, size_t ws_size,
                              hipStream_t stream) {
  const float* x    = (const float*)d_in[0];
  const int*   ei   = (const int*)  d_in[1];
  const float* Win  = (const float*)d_in[2];
  const float* bin  = (const float*)d_in[3];
  const float* Wl   = (const float*)d_in[4];
  const float* Wout = (const float*)d_in[5];
  const float* bout = (const float*)d_in[6];
  float* out = (float*)d_out;

  float* ws   = (float*)d_ws;
  float* dinv = ws;                                // N
  float* h    = ws + NN;                           // N*H
  float* h0   = h + (size_t)NN * HH;               // N*H
  float* agg  = h0 + (size_t)NN * HH;              // N*H

  const int ew = NN * HH / 4;                      // float4 elements

  k_init_deg     <<<(NN + 255) / 256, 256, 0, stream>>>(dinv);
  k_accum_deg    <<<(EE + 255) / 256, 256, 0, stream>>>(ei, dinv);
  k_finalize_dinv<<<(NN + 255) / 256, 256, 0, stream>>>(dinv);

  k_gemm_in<<<NN / 16, 256, 0, stream>>>(x, Win, bin, h, h0);

  for (int l = 0; l < LL; ++l) {
    float beta = logf(0.5f / (float)(l + 1) + 1.0f);
    k_agg_init    <<<(ew + 255) / 256, 256, 0, stream>>>(h, dinv, agg);
    k_edge_scatter<<<(EE * 32 + 255) / 256, 256, 0, stream>>>(ei, dinv, h, agg);
    k_support     <<<(ew + 255) / 256, 256, 0, stream>>>(agg, h0);
    k_gemm_layer  <<<NN / 16, 256, 0, stream>>>(agg, Wl + (size_t)l * HH * HH, h, beta);
  }

  k_gemm_out<<<NN / 16, 128, 0, stream>>>(h, Wout, bout, out);
}